// ConstrainedProbabilityMatrixFactorization_7576322310166
// MI455X (gfx1250) — compile-verified
//
#include <hip/hip_runtime.h>
#include <hip/hip_bf16.h>

#define N_BU 1024
#define N_BI 4096
#define DIM  64
#define LNZ  50

typedef __attribute__((ext_vector_type(2))) float v2f;
typedef __attribute__((ext_vector_type(8))) float v8f;

// ---------------------------------------------------------------------------
// Phase 1: A[b][d] = user_weight[uid[b]][d] + sum_l fb_values[b][l] * irew[fb_idx[b][l]][d]
// One block (64 threads = 2 waves) per user row; d = threadIdx.x.
// fb index/value loads are wave-uniform (broadcast from L0/L2).
// ---------------------------------------------------------------------------
__global__ __launch_bounds__(64)
void prep_users(const int* __restrict__ user_ids,
                const int* __restrict__ fb_indices,
                const float* __restrict__ fb_values,
                const float* __restrict__ user_weight,
                const float* __restrict__ user_bias,
                const float* __restrict__ irew,
                float* __restrict__ Aout,   // [N_BU][DIM]
                float* __restrict__ ub)     // [N_BU]
{
    const int b = blockIdx.x;
    const int d = threadIdx.x;
    const int uid = user_ids[b];
    float acc = user_weight[(size_t)uid * DIM + d];
    const int*   idxr = fb_indices + b * LNZ;
    const float* valr = fb_values  + b * LNZ;
#pragma unroll 2
    for (int l = 0; l < LNZ; ++l) {
        const int   it = idxr[l];
        const float v  = valr[l];
        acc = fmaf(v, irew[(size_t)it * DIM + d], acc);
    }
    Aout[b * DIM + d] = acc;
    if (d == 0) ub[b] = user_bias[uid];
}

// ---------------------------------------------------------------------------
// Phase 2: stage B = item_weight[item_ids].T in K-pair-interleaved layout:
//   Bt2[p][n][e] = item_weight[item_ids[n]][2*p + e],  p in [0,32), e in {0,1}
// so a WMMA B fragment is one aligned float2 per lane, coalesced over n.
// ---------------------------------------------------------------------------
__global__ __launch_bounds__(256)
void prep_items(const int* __restrict__ item_ids,
                const float* __restrict__ item_weight,
                const float* __restrict__ item_bias,
                float* __restrict__ Bt2,    // [DIM/2][N_BI][2]
                float* __restrict__ ib)     // [N_BI]
{
    const int t = blockIdx.x * 256 + threadIdx.x;   // 0 .. N_BI*DIM-1
    const int n = t >> 6;
    const int k = t & 63;
    const int iid = item_ids[n];
    const float w = item_weight[(size_t)iid * DIM + k];
    Bt2[((size_t)(k >> 1) * N_BI + n) * 2 + (k & 1)] = w;
    if (k == 0) ib[n] = item_bias[iid];
}

// ---------------------------------------------------------------------------
// Phase 3: rating = A @ Bt + ub + ib^T + bias via V_WMMA_F32_16X16X4_F32.
// One wave per 16(M) x 64(N) output tile; 4 f32 accumulators (v8f each),
// 16 K-steps of K=4. A-fragment layout (ISA 32-bit 16x4 A):
//   lane = M + 16*(K>=2), VGPR = K%2  ->  float2 {A[m][4k+2h], A[m][4k+2h+1]}
// B-fragment mirrors it over N. C/D layout: VGPR v, half h -> M=8h+v, N=lane&15.
// ---------------------------------------------------------------------------
__global__ __launch_bounds__(256)
void gemm_rating(const float* __restrict__ A,     // [N_BU][DIM]
                 const float* __restrict__ Bt2,   // [DIM/2][N_BI][2]
                 const float* __restrict__ ub,    // [N_BU]
                 const float* __restrict__ ib,    // [N_BI]
                 const float* __restrict__ biasp, // [1]
                 float* __restrict__ out)         // [N_BU][N_BI]
{
    const int lane  = threadIdx.x & 31;
    const int wave  = threadIdx.x >> 5;
    const int tile  = blockIdx.x * 8 + wave;      // 64 M-tiles x 64 N-strips
    const int m0    = (tile >> 6) << 4;           // 0..1008
    const int n0    = (tile & 63) << 6;           // 0..4032
    const int mrow  = lane & 15;                  // M (for A) / N (for B)
    const int khalf = lane >> 4;                  // selects K pair within K=4

    v8f acc0 = {}, acc1 = {}, acc2 = {}, acc3 = {};

    const float* arow  = A   + (m0 + mrow) * DIM + khalf * 2;
    const float* bbase = Bt2 + (size_t)khalf * (2 * N_BI) + (size_t)(n0 + mrow) * 2;

#pragma unroll 4
    for (int k0 = 0; k0 < 16; ++k0) {
        const v2f a = *(const v2f*)(arow + k0 * 4);
        const float* brow = bbase + (size_t)k0 * (4 * N_BI);
        const v2f b0 = *(const v2f*)(brow);
        const v2f b1 = *(const v2f*)(brow + 32);
        const v2f b2 = *(const v2f*)(brow + 64);
        const v2f b3 = *(const v2f*)(brow + 96);
        acc0 = __builtin_amdgcn_wmma_f32_16x16x4_f32(false, a, false, b0, (short)0, acc0, false, false);
        acc1 = __builtin_amdgcn_wmma_f32_16x16x4_f32(false, a, false, b1, (short)0, acc1, false, false);
        acc2 = __builtin_amdgcn_wmma_f32_16x16x4_f32(false, a, false, b2, (short)0, acc2, false, false);
        acc3 = __builtin_amdgcn_wmma_f32_16x16x4_f32(false, a, false, b3, (short)0, acc3, false, false);
    }

    const float biasv = biasp[0];
    const int ncol = mrow;
    const float ib0 = ib[n0 + ncol]      + biasv;
    const float ib1 = ib[n0 + 16 + ncol] + biasv;
    const float ib2 = ib[n0 + 32 + ncol] + biasv;
    const float ib3 = ib[n0 + 48 + ncol] + biasv;

#pragma unroll
    for (int v = 0; v < 8; ++v) {
        const int row = m0 + khalf * 8 + v;
        const float ubv = ub[row];
        float* orow = out + (size_t)row * N_BI + n0 + ncol;
        orow[0]  = acc0[v] + ubv + ib0;
        orow[16] = acc1[v] + ubv + ib1;
        orow[32] = acc2[v] + ubv + ib2;
        orow[48] = acc3[v] + ubv + ib3;
    }
}

// ---------------------------------------------------------------------------
extern "C" void kernel_launch(void* const* d_in, const int* in_sizes, int n_in,
                              void* d_out, int out_size, void* d_ws, size_t ws_size,
                              hipStream_t stream) {
    (void)in_sizes; (void)n_in; (void)out_size; (void)ws_size;

    const int*   user_ids    = (const int*)  d_in[0];
    const int*   item_ids    = (const int*)  d_in[1];
    const int*   fb_indices  = (const int*)  d_in[2];
    const float* fb_values   = (const float*)d_in[3];
    const float* user_weight = (const float*)d_in[4];
    const float* user_bias   = (const float*)d_in[5];
    const float* item_weight = (const float*)d_in[6];
    const float* item_bias   = (const float*)d_in[7];
    const float* bias        = (const float*)d_in[8];
    const float* irew        = (const float*)d_in[9];
    float* out = (float*)d_out;

    // Workspace layout (fp32): A[1024*64] | Bt2[64*4096] | ub[1024] | ib[4096]
    float* A   = (float*)d_ws;
    float* Bt2 = A   + N_BU * DIM;
    float* ub  = Bt2 + DIM * N_BI;
    float* ib  = ub  + N_BU;

    prep_users<<<N_BU, 64, 0, stream>>>(user_ids, fb_indices, fb_values,
                                        user_weight, user_bias, irew, A, ub);
    prep_items<<<(N_BI * DIM) / 256, 256, 0, stream>>>(item_ids, item_weight,
                                                       item_bias, Bt2, ib);
    // 64 M-tiles * 64 N-strips = 4096 waves, 8 waves/block -> 512 blocks
    gemm_rating<<<(N_BU / 16) * (N_BI / 64) / 8, 256, 0, stream>>>(A, Bt2, ub, ib,
                                                                   bias, out);
}